// cbow_2697239462020
// MI455X (gfx1250) — compile-verified
//
#include <hip/hip_runtime.h>
#include <hip/hip_bf16.h>

// CBOW: gather-sum (emb, padding_idx=0) then GEMM out = summed @ W^T + b
// M=4096 (batch), N=32000 (vocab), K=128 (embed). f32 WMMA (16x16x4) keeps
// full reference precision; kernel is output-store bound (~524MB @ 23.3TB/s).
// LDS staging uses GLOBAL_LOAD_ASYNC_TO_LDS_B128 (ASYNCcnt path): builtin
// signature per compiler diagnostic takes a generic `int vector_size(16)*`
// source; LDS dest passed as addrspace(3)-qualified pointer.

typedef __attribute__((ext_vector_type(2))) float v2f;
typedef __attribute__((ext_vector_type(8))) float v8f;

// Exact element type the async builtin expects (per hipcc diagnostic).
typedef int v4i __attribute__((vector_size(4 * sizeof(int))));
typedef __attribute__((address_space(3))) v4i* lds_v4i_p;

#if defined(__AMDGCN__) && __has_builtin(__builtin_amdgcn_global_load_async_to_lds_b128)
#define HAVE_ASYNC_LDS 1
#else
#define HAVE_ASYNC_LDS 0
#endif

#define BATCH 4096
#define CTX   8
#define EMBED 128
#define VOCAB 32000

#define BM  128
#define BN  128
#define LDK 132   // 128 + 4 pad: lane stride 132 -> banks advance 4/row, no conflicts

__device__ __forceinline__ void wait_async_lds() {
#if HAVE_ASYNC_LDS
#if __has_builtin(__builtin_amdgcn_s_wait_asynccnt)
    __builtin_amdgcn_s_wait_asynccnt(0);
#else
    asm volatile("s_wait_asynccnt 0" ::: "memory");
#endif
#endif
}

// ---------------- Kernel 1: CBOW gather-sum (one wave per sample) -----------
__global__ __launch_bounds__(256) void cbow_gather(const int* __restrict__ x,
                                                   const float* __restrict__ emb,
                                                   float* __restrict__ summed) {
    const int sample = (blockIdx.x * blockDim.x + threadIdx.x) >> 5;
    const int lane   = threadIdx.x & 31;
    if (sample >= BATCH) return;

    float4 acc = make_float4(0.f, 0.f, 0.f, 0.f);
    const int* xs = x + sample * CTX;
#pragma unroll
    for (int c = 0; c < CTX; ++c) {
        const int idx = xs[c];
        if (idx != 0) {  // padding_idx=0 contributes zero
            const float4 e = *(const float4*)(emb + (size_t)idx * EMBED + lane * 4);
            acc.x += e.x; acc.y += e.y; acc.z += e.z; acc.w += e.w;
        }
    }
    *(float4*)(summed + (size_t)sample * EMBED + lane * 4) = acc;
}

// ---------------- Kernel 2: WMMA GEMM, 128x128 block, K=128 single pass -----
__global__ __launch_bounds__(256) void cbow_gemm(const float* __restrict__ S,
                                                 const float* __restrict__ W,
                                                 const float* __restrict__ bias,
                                                 float* __restrict__ out) {
    __shared__ __align__(16) float sA[BM * LDK];   // summed block, row-major [m][k]
    __shared__ __align__(16) float sB[BN * LDK];   // W block,      row-major [n][k]

    const int n0 = blockIdx.x * BN;
    const int m0 = blockIdx.y * BM;
    const int t  = threadIdx.x;

    // Stage A and B blocks: 128 rows x 128 cols each, one B128 per slot.
    // 4096 slots / 256 threads = 16 iterations per matrix.
#pragma unroll
    for (int i = 0; i < 16; ++i) {
        const int li  = t + i * 256;       // 0..4095
        const int row = li >> 5;           // 0..127
        const int c4  = (li & 31) << 2;    // k offset in floats (multiple of 4)
#if HAVE_ASYNC_LDS
        // Memory -> LDS directly, tracked on ASYNCcnt (no VGPR round trip).
        __builtin_amdgcn_global_load_async_to_lds_b128(
            (v4i*)(S + (size_t)(m0 + row) * EMBED + c4),
            (lds_v4i_p)(unsigned)(uintptr_t)(const void*)(sA + row * LDK + c4),
            0, 0);
        __builtin_amdgcn_global_load_async_to_lds_b128(
            (v4i*)(W + (size_t)(n0 + row) * EMBED + c4),
            (lds_v4i_p)(unsigned)(uintptr_t)(const void*)(sB + row * LDK + c4),
            0, 0);
#else
        const float4 va = *(const float4*)(S + (size_t)(m0 + row) * EMBED + c4);
        *(float4*)(sA + row * LDK + c4) = va;
        const float4 vb = *(const float4*)(W + (size_t)(n0 + row) * EMBED + c4);
        *(float4*)(sB + row * LDK + c4) = vb;
#endif
    }
    wait_async_lds();   // ASYNCcnt==0: all LDS writes of this wave landed
    __syncthreads();

    const int lane = t & 31;
    const int wave = t >> 5;            // 8 waves
    const int wMo  = (wave >> 2) * 64;  // wave M origin in block (0 or 64)
    const int wNo  = (wave & 3) * 32;   // wave N origin in block (0..96)
    const int kh   = lane >> 4;         // lane group: K half (0/1), C row half
    const int l16  = lane & 15;

    v8f acc[4][2];
#pragma unroll
    for (int mi = 0; mi < 4; ++mi)
#pragma unroll
        for (int ni = 0; ni < 2; ++ni)
            acc[mi][ni] = 0.f;

    // K loop: 32 steps of k+=4, 8 WMMAs per step (4 M-tiles x 2 N-tiles).
#pragma unroll 4
    for (int k0 = 0; k0 < EMBED; k0 += 4) {
        const int kk = k0 + (kh << 1);  // A/B frag: VGPR v holds K = kk + v
        v2f afr[4], bfr[2];
#pragma unroll
        for (int mi = 0; mi < 4; ++mi)
            afr[mi] = *(const v2f*)(sA + (wMo + mi * 16 + l16) * LDK + kk);
#pragma unroll
        for (int ni = 0; ni < 2; ++ni)
            bfr[ni] = *(const v2f*)(sB + (wNo + ni * 16 + l16) * LDK + kk);
#pragma unroll
        for (int mi = 0; mi < 4; ++mi)
#pragma unroll
            for (int ni = 0; ni < 2; ++ni)
                acc[mi][ni] = __builtin_amdgcn_wmma_f32_16x16x4_f32(
                    false, afr[mi], false, bfr[ni], (short)0, acc[mi][ni],
                    false, false);
    }

    // Bias add + store. C/D layout: lane -> N = l16, VGPR v -> M = kh*8 + v.
#pragma unroll
    for (int ni = 0; ni < 2; ++ni) {
        const int n  = n0 + wNo + ni * 16 + l16;
        const float bv = bias[n];
#pragma unroll
        for (int mi = 0; mi < 4; ++mi) {
            const int rbase = m0 + wMo + mi * 16 + kh * 8;
#pragma unroll
            for (int v = 0; v < 8; ++v) {
                out[(size_t)(rbase + v) * VOCAB + n] = acc[mi][ni][v] + bv;
            }
        }
    }
}

extern "C" void kernel_launch(void* const* d_in, const int* in_sizes, int n_in,
                              void* d_out, int out_size, void* d_ws, size_t ws_size,
                              hipStream_t stream) {
    const int*   x   = (const int*)d_in[0];     // [4096, 8] (integer -> int32 per harness)
    const float* emb = (const float*)d_in[1];   // [32000, 128]
    const float* W   = (const float*)d_in[2];   // [32000, 128]
    const float* b   = (const float*)d_in[3];   // [32000]
    float* out    = (float*)d_out;              // [4096, 32000]
    float* summed = (float*)d_ws;               // [4096, 128] scratch (2 MB)

    // Kernel 1: 4096 samples, one wave each -> 4096*32 threads.
    cbow_gather<<<(BATCH * 32) / 256, 256, 0, stream>>>(x, emb, summed);

    // Kernel 2: grid (N/BN, M/BM) = (250, 32), 256 threads (8 waves).
    dim3 grid(VOCAB / BN, BATCH / BM);
    cbow_gemm<<<grid, 256, 0, stream>>>(summed, W, b, out);
}